// OPID_78769700208710
// MI455X (gfx1250) — compile-verified
//
#include <hip/hip_runtime.h>
#include <hip/hip_bf16.h>
#include <math.h>

#define N_NODES 20000
#define BATCH   64
#define NUM_STEPS 6
#define NUM_REL 6
#define EDGES   400000
#define HID     64

typedef __attribute__((ext_vector_type(2)))  float v2f;
typedef __attribute__((ext_vector_type(8)))  float v8f;

#if defined(__has_builtin)
#if __has_builtin(__builtin_amdgcn_wmma_f32_16x16x4_f32)
#define HAVE_WMMA_F32X4 1
#endif
#endif

// ---------------------------------------------------------------------------
// Scalar prep: coef[r] = sign[r]*softplus(g_logits[r]); alphas[k] = sigmoid;
// ceW2[b] = dot(cell_emb[cell_idx[b]], W2)   (decode constant per batch row)
// ---------------------------------------------------------------------------
__global__ void setup_kernel(const float* __restrict__ g_logits,
                             const float* __restrict__ alpha_logits,
                             const float* __restrict__ cell_emb,
                             const int*   __restrict__ cell_idx,
                             const float* __restrict__ W2,
                             float* __restrict__ coef,
                             float* __restrict__ alphas,
                             float* __restrict__ ceW2) {
  int t = threadIdx.x;
  const float SIGNS[NUM_REL] = {1.f, -1.f, 1.f, -1.f, 1.f, -1.f};
  if (t < NUM_REL) {
    float x = g_logits[t];
    float sp = (x > 20.f) ? x : log1pf(expf(x));
    coef[t] = SIGNS[t] * sp;
  }
  if (t < NUM_STEPS) {
    float x = alpha_logits[t];
    alphas[t] = 1.f / (1.f + expf(-x));
  }
  if (t < BATCH) {
    const float* ce = cell_emb + (size_t)cell_idx[t] * HID;
    float s = 0.f;
#pragma unroll
    for (int h = 0; h < HID; ++h) s += ce[h] * W2[h];
    ceW2[t] = s;
  }
}

// ---------------------------------------------------------------------------
// Transpose u_raw [B,N] -> node-major [N,B] via padded LDS tile (64x65).
// Writes both hT (working state) and h0T (residual source).
// ---------------------------------------------------------------------------
__global__ void transpose_kernel(const float* __restrict__ u_raw,
                                 float* __restrict__ hT,
                                 float* __restrict__ h0T) {
  __shared__ float tile[64][65];
  int n0 = blockIdx.x * 64;
  int t = threadIdx.x; // 256 threads
#pragma unroll
  for (int i = 0; i < 16; ++i) {
    int idx = t + i * 256;
    int bb = idx >> 6, nn = idx & 63;
    int n = n0 + nn;
    tile[bb][nn] = (n < N_NODES) ? u_raw[(size_t)bb * N_NODES + n] : 0.f;
  }
  __syncthreads();
#pragma unroll
  for (int i = 0; i < 16; ++i) {
    int idx = t + i * 256;
    int nn = idx >> 6, bb = idx & 63;
    int n = n0 + nn;
    if (n < N_NODES) {
      float v = tile[bb][nn];
      hT [(size_t)n * BATCH + bb] = v;
      h0T[(size_t)n * BATCH + bb] = v;
    }
  }
}

__global__ void zero_kernel(float* __restrict__ p, int n) {
  int i = blockIdx.x * blockDim.x + threadIdx.x;
  if (i < n) p[i] = 0.f;
}

// ---------------------------------------------------------------------------
// One wave per edge (all 6 relations fused, index = r*E + e).
// Coalesced 256B gather of hT[src][:], 64 f32 atomic adds into msgT[dst][:].
// Whole state + edge lists are L2-resident (192MB); atomics resolve in L2.
// ---------------------------------------------------------------------------
__global__ void edge_kernel(const float* __restrict__ hT,
                            float*       __restrict__ msgT,
                            const int*   __restrict__ edge_src,
                            const int*   __restrict__ edge_dst,
                            const float* __restrict__ edge_val,
                            const float* __restrict__ coef) {
  long long wave = ((long long)blockIdx.x * blockDim.x + threadIdx.x) >> 5;
  int lane = threadIdx.x & 31;
  if (wave >= (long long)NUM_REL * EDGES) return;
  int r = (int)(wave / EDGES);
  float c = coef[r] * edge_val[wave];
  int src = edge_src[wave];
  int dst = edge_dst[wave];
  const float2* hp = (const float2*)(hT + (size_t)src * BATCH);
  float2 hv = hp[lane];
  float* mp = msgT + (size_t)dst * BATCH + lane * 2;
  atomicAdd(mp,     c * hv.x);
  atomicAdd(mp + 1, c * hv.y);
}

// h = a*h0 + (1-a)*msg; re-zero msg for the next step in the same pass.
__global__ void update_kernel(float* __restrict__ hT,
                              float* __restrict__ msgT,
                              const float* __restrict__ h0T,
                              const float* __restrict__ alphas, int k) {
  int i = blockIdx.x * blockDim.x + threadIdx.x;
  float a = alphas[k];
  float m = msgT[i];
  msgT[i] = 0.f;
  hT[i] = a * h0T[i] + (1.f - a) * m;
}

// ---------------------------------------------------------------------------
// Decode: per wave, 16 (b,n) elements. D = A(16x4) x B(4x16) with
// V_WMMA_F32_16X16X4_F32; 4 tiles cover H=64. A layout (32-bit, 16x4):
// lanes 0-15 hold K=0,1 (ctl,u); lanes 16-31 hold K=2,3 (h,0).
// B layout mirrors: VGPR0 = rows {0,2}, VGPR1 = rows {1,3} of W1 (row3 = 0).
// D: vgpr v, lanes<16 -> M=v, lanes>=16 -> M=v+8; N = lane&15 = hidden idx.
// All loads are unconditional (every address valid for every lane); the
// half-wave selection is pure v_cndmask, so the 4 WMMAs issue branch-free.
// ---------------------------------------------------------------------------
#ifdef HAVE_WMMA_F32X4
__global__ void decode_kernel(const float* __restrict__ ctl,
                              const float* __restrict__ u_raw,
                              const float* __restrict__ hT,
                              const float* __restrict__ W1,
                              const float* __restrict__ b1,
                              const float* __restrict__ W2,
                              const float* __restrict__ b2,
                              const float* __restrict__ ceW2,
                              float* __restrict__ out) {
  int wave = (blockIdx.x * blockDim.x + threadIdx.x) >> 5;
  int lane = threadIdx.x & 31;
  int b  = wave / (N_NODES / 16);
  int n0 = (wave % (N_NODES / 16)) * 16;
  int m  = lane & 15;
  bool hi = lane >= 16;

  // --- unconditional loads, then lane-half select (no exec branching) ---
  float a_ctl = ctl  [(size_t)b * N_NODES + n0 + m];
  float a_u   = u_raw[(size_t)b * N_NODES + n0 + m];
  float a_h   = hT[(size_t)(n0 + m) * BATCH + b];

  float w1r0[4], w1r1[4], w1r2[4], bias[4], w2[4];
#pragma unroll
  for (int t = 0; t < 4; ++t) {
    int h = t * 16 + m;
    w1r0[t] = W1[0 * HID + h];
    w1r1[t] = W1[1 * HID + h];
    w1r2[t] = W1[2 * HID + h];
    bias[t] = b1[h];
    w2[t]   = W2[h];
  }

  v2f A;
  A.x = hi ? a_h : a_ctl;   // K=2 : K=0
  A.y = hi ? 0.f : a_u;     // K=3(pad) : K=1

  float acc[8];
#pragma unroll
  for (int v = 0; v < 8; ++v) acc[v] = 0.f;

#pragma unroll
  for (int t = 0; t < 4; ++t) {
    v2f Bm;
    Bm.x = hi ? w1r2[t] : w1r0[t];  // rows 2 / 0
    Bm.y = hi ? 0.f     : w1r1[t];  // rows 3(pad) / 1
    v8f C = {};
    v8f D = __builtin_amdgcn_wmma_f32_16x16x4_f32(
        false, A, false, Bm, (short)0, C, false, false);
#pragma unroll
    for (int v = 0; v < 8; ++v) {
      float z = D[v] + bias[t];
      z = z > 0.f ? z : 0.f;
      acc[v] += z * w2[t];
    }
  }
  // reduce hidden dim across the 16 lanes of each half (masks 1..8 stay local)
#pragma unroll
  for (int off = 1; off < 16; off <<= 1)
#pragma unroll
    for (int v = 0; v < 8; ++v)
      acc[v] += __shfl_xor(acc[v], off, 32);

  if (m == 0) {
    float add = ceW2[b] + b2[0];
    int mOff = hi ? 8 : 0;
#pragma unroll
    for (int v = 0; v < 8; ++v)
      out[(size_t)b * N_NODES + n0 + mOff + v] = acc[v] + add;
  }
}
#else
// Fallback (builtin unavailable): plain VALU decode, one thread per (b,n).
__global__ void decode_kernel(const float* __restrict__ ctl,
                              const float* __restrict__ u_raw,
                              const float* __restrict__ hT,
                              const float* __restrict__ W1,
                              const float* __restrict__ b1,
                              const float* __restrict__ W2,
                              const float* __restrict__ b2,
                              const float* __restrict__ ceW2,
                              float* __restrict__ out) {
  int i = blockIdx.x * blockDim.x + threadIdx.x;
  int b = i / N_NODES, n = i % N_NODES;
  float x0 = ctl[i], x1 = u_raw[i], x2 = hT[(size_t)n * BATCH + b];
  float y = ceW2[b] + b2[0];
  for (int h = 0; h < HID; ++h) {
    float z = x0 * W1[h] + x1 * W1[HID + h] + x2 * W1[2 * HID + h] + b1[h];
    y += (z > 0.f ? z : 0.f) * W2[h];
  }
  out[i] = y;
}
#endif

extern "C" void kernel_launch(void* const* d_in, const int* in_sizes, int n_in,
                              void* d_out, int out_size, void* d_ws, size_t ws_size,
                              hipStream_t stream) {
  const float* ctl_base     = (const float*)d_in[0];
  const float* u_raw        = (const float*)d_in[1];
  const int*   cell_idx     = (const int*)  d_in[2];
  const int*   edge_src     = (const int*)  d_in[3];
  const int*   edge_dst     = (const int*)  d_in[4];
  const float* edge_val     = (const float*)d_in[5];
  const float* g_logits     = (const float*)d_in[6];
  const float* alpha_logits = (const float*)d_in[7];
  const float* cell_emb     = (const float*)d_in[8];
  const float* W1           = (const float*)d_in[9];
  const float* b1           = (const float*)d_in[10];
  const float* W2           = (const float*)d_in[11];
  const float* b2           = (const float*)d_in[12];
  float* out = (float*)d_out;

  const size_t NB = (size_t)N_NODES * BATCH; // 1,280,000
  float* hT     = (float*)d_ws;
  float* h0T    = hT  + NB;
  float* msgT   = h0T + NB;
  float* coef   = msgT + NB;   // 6
  float* alphas = coef + 8;    // 6
  float* ceW2   = alphas + 8;  // 64

  setup_kernel<<<1, 64, 0, stream>>>(g_logits, alpha_logits, cell_emb,
                                     cell_idx, W2, coef, alphas, ceW2);
  transpose_kernel<<<(N_NODES + 63) / 64, 256, 0, stream>>>(u_raw, hT, h0T);
  zero_kernel<<<(int)(NB / 256), 256, 0, stream>>>(msgT, (int)NB);

  long long totalEdges = (long long)NUM_REL * EDGES;          // 2.4M edges
  int edgeBlocks = (int)((totalEdges * 32 + 255) / 256);      // 1 wave/edge
  for (int k = 0; k < NUM_STEPS; ++k) {
    edge_kernel<<<edgeBlocks, 256, 0, stream>>>(hT, msgT, edge_src, edge_dst,
                                                edge_val, coef);
    update_kernel<<<(int)(NB / 256), 256, 0, stream>>>(hT, msgT, h0T, alphas, k);
  }

#ifdef HAVE_WMMA_F32X4
  // 80,000 full waves (exact): EXEC all-ones at every WMMA.
  int decodeBlocks = (BATCH * (N_NODES / 16)) / 8; // 8 waves per 256-thr block
  decode_kernel<<<decodeBlocks, 256, 0, stream>>>(ctl_base, u_raw, hT, W1, b1,
                                                  W2, b2, ceW2, out);
#else
  decode_kernel<<<(int)(NB / 256), 256, 0, stream>>>(ctl_base, u_raw, hT, W1,
                                                     b1, W2, b2, ceW2, out);
#endif
}